// SubMConv2dTestTorch_8942121910439
// MI455X (gfx1250) — compile-verified
//
#include <hip/hip_runtime.h>

// Problem constants (match reference)
#define BSZ  4
#define HDIM 512
#define WDIM 512
#define CINC 64
#define COUTC 64
#define NTAP 9
#define WSTRIDE 80   // padded LDS row stride (floats): 64 data + 16 pad -> conflict-free

typedef __attribute__((ext_vector_type(2))) float v2f;
typedef __attribute__((ext_vector_type(8))) float v8f;
typedef __attribute__((ext_vector_type(4))) unsigned int u32x4;
typedef __attribute__((ext_vector_type(4))) int i32x4;
typedef __attribute__((ext_vector_type(8))) int i32x8;

__global__ void init_grid_kernel(int* __restrict__ grid, int total) {
    int i = blockIdx.x * blockDim.x + threadIdx.x;
    if (i < total) grid[i] = -1;
}

__global__ void scatter_grid_kernel(const int* __restrict__ coors,
                                    int* __restrict__ grid, int n) {
    int i = blockIdx.x * blockDim.x + threadIdx.x;
    if (i < n) {
        int b = coors[3 * i + 0];
        int y = coors[3 * i + 1];
        int x = coors[3 * i + 2];
        grid[(b * HDIM + y) * WDIM + x] = i;
    }
}

// nbr[k*n + i] = row index of neighbor of point i at tap k, or -1
__global__ void build_nbr_kernel(const int* __restrict__ coors,
                                 const int* __restrict__ grid,
                                 int* __restrict__ nbr, int n) {
    int i = blockIdx.x * blockDim.x + threadIdx.x;
    if (i >= n) return;
    int b = coors[3 * i + 0];
    int y = coors[3 * i + 1];
    int x = coors[3 * i + 2];
    int k = 0;
    #pragma unroll
    for (int dy = -1; dy <= 1; ++dy) {
        #pragma unroll
        for (int dx = -1; dx <= 1; ++dx) {
            int ny = y + dy;
            int nx = x + dx;
            int idx = -1;
            if (ny >= 0 && ny < HDIM && nx >= 0 && nx < WDIM)
                idx = grid[(b * HDIM + ny) * WDIM + nx];
            nbr[k * n + i] = idx;
            ++k;
        }
    }
}

// TDM: DMA one 64x64 f32 weight tile (16 KB) into LDS with row padding 64->80
// floats (pad_interval = 64 DWORDs, pad_amount = 16 DWORDs).
__device__ inline void tdm_load_tap(const float* gsrc, unsigned lds_byte_off) {
    unsigned long long ga = (unsigned long long)(uintptr_t)gsrc;
    u32x4 g0;
    g0[0] = 1u;                                         // count=1 (valid, user)
    g0[1] = lds_byte_off;                               // LDS dest byte address
    g0[2] = (unsigned)(ga & 0xffffffffu);               // global_addr[31:0]
    g0[3] = (unsigned)((ga >> 32) & 0x1ffffffu)         // global_addr[56:32]
            | (2u << 30);                               // type = 2 ("image")
    i32x8 g1;
    g1[0] = (int)((2u << 16)       // data_size = 4 bytes
                | (1u << 20)       // pad_enable
                | (5u << 22)       // pad_interval: 2^(5+1) = 64 DWORDs
                | (15u << 25));    // pad_amount: 15+1 = 16 DWORDs
    g1[1] = 64 << 16;              // tensor_dim0 = 64 (bits 79:48)
    g1[2] = 64 << 16;              // tensor_dim1 = 64 (bits 111:80)
    g1[3] = 64 << 16;              // tile_dim0  = 64 (bits 127:112)
    g1[4] = 64;                    // tile_dim1  = 64 (bits 143:128)
    g1[5] = 64;                    // tensor_dim0_stride = 64 (bits 207:160)
    g1[6] = 0;
    g1[7] = 0;
    i32x4 z4 = {0, 0, 0, 0};
#if __has_include(<hip/amd_detail/amd_gfx1250_TDM.h>)
    i32x8 z8 = {0, 0, 0, 0, 0, 0, 0, 0};
    __builtin_amdgcn_tensor_load_to_lds(g0, g1, z4, z4, z8, 0);
#else
    __builtin_amdgcn_tensor_load_to_lds(g0, g1, z4, z4, 0);
#endif
}

// One wave computes a 32x64 output tile via V_WMMA_F32_16X16X4_F32.
// Block = 256 threads = 8 waves = 256 rows. Weights double-buffered in LDS,
// staged one tap ahead by the Tensor Data Mover.
__launch_bounds__(256)
__global__ void subm_conv_wmma(const float* __restrict__ feats,
                               const float* __restrict__ w,   // [9][64][64]
                               const int*  __restrict__ nbr,  // [9][n]
                               float* __restrict__ out, int n) {
    __shared__ float wbuf[2 * CINC * WSTRIDE];   // 2 x 20 KB

    const int tid   = threadIdx.x;
    const int lane  = tid & 31;
    const int wave  = tid >> 5;
    const int rowbase = blockIdx.x * 256 + wave * 32;

    const int mlane = lane & 15;      // M row within 16-tile / N col within tile
    const int khalf = lane >> 4;      // selects K pair {0,1} vs {2,3}

    const int mrow0 = rowbase + mlane;        // rows 0..15 of wave tile
    const int mrow1 = rowbase + 16 + mlane;   // rows 16..31 of wave tile

    const unsigned lds_base = (unsigned)(uintptr_t)(&wbuf[0]);
    const unsigned BUFBYTES = CINC * WSTRIDE * 4u;   // 20480

    v8f accA0 = {}, accA1 = {}, accA2 = {}, accA3 = {};
    v8f accB0 = {}, accB1 = {}, accB2 = {}, accB3 = {};

    // Prologue: DMA tap 0 weights into buffer 0.
    if (wave == 0) tdm_load_tap(w, lds_base);

    for (int k = 0; k < NTAP; ++k) {
        __builtin_amdgcn_s_wait_tensorcnt(0);   // tap k's DMA complete (wave 0)
        __syncthreads();                        // publish buf[k&1]; retire buf[(k+1)&1]
        if (wave == 0 && (k + 1) < NTAP)        // overlap next tap's DMA with compute
            tdm_load_tap(w + (size_t)(k + 1) * CINC * COUTC,
                         lds_base + (unsigned)((k + 1) & 1) * BUFBYTES);

        const float* wb = wbuf + (size_t)(k & 1) * (CINC * WSTRIDE);

        int idx0 = (mrow0 < n) ? nbr[k * n + mrow0] : -1;
        int idx1 = (mrow1 < n) ? nbr[k * n + mrow1] : -1;
        const bool v0 = idx0 >= 0;
        const bool v1 = idx1 >= 0;
        const float* arow0 = feats + (size_t)(v0 ? idx0 : 0) * CINC + (khalf << 1);
        const float* arow1 = feats + (size_t)(v1 ? idx1 : 0) * CINC + (khalf << 1);
        const float* brow  = wb + (size_t)(khalf << 1) * WSTRIDE + mlane;

        #pragma unroll
        for (int kk = 0; kk < 16; ++kk) {
            v2f a0 = {}, a1 = {};
            if (v0) a0 = *(const v2f*)(arow0 + kk * 4);   // predicated 8B gathers
            if (v1) a1 = *(const v2f*)(arow1 + kk * 4);

            const float* bp = brow + (size_t)(kk * 4) * WSTRIDE;
            v2f b0, b1, b2, b3;
            b0[0] = bp[ 0];  b0[1] = bp[ 0 + WSTRIDE];
            b1[0] = bp[16];  b1[1] = bp[16 + WSTRIDE];
            b2[0] = bp[32];  b2[1] = bp[32 + WSTRIDE];
            b3[0] = bp[48];  b3[1] = bp[48 + WSTRIDE];

            accA0 = __builtin_amdgcn_wmma_f32_16x16x4_f32(false, a0, false, b0,
                                                          (short)0, accA0, false, false);
            accB0 = __builtin_amdgcn_wmma_f32_16x16x4_f32(false, a1, false, b0,
                                                          (short)0, accB0, false, false);
            accA1 = __builtin_amdgcn_wmma_f32_16x16x4_f32(false, a0, false, b1,
                                                          (short)0, accA1, false, false);
            accB1 = __builtin_amdgcn_wmma_f32_16x16x4_f32(false, a1, false, b1,
                                                          (short)0, accB1, false, false);
            accA2 = __builtin_amdgcn_wmma_f32_16x16x4_f32(false, a0, false, b2,
                                                          (short)0, accA2, false, false);
            accB2 = __builtin_amdgcn_wmma_f32_16x16x4_f32(false, a1, false, b2,
                                                          (short)0, accB2, false, false);
            accA3 = __builtin_amdgcn_wmma_f32_16x16x4_f32(false, a0, false, b3,
                                                          (short)0, accA3, false, false);
            accB3 = __builtin_amdgcn_wmma_f32_16x16x4_f32(false, a1, false, b3,
                                                          (short)0, accB3, false, false);
        }
    }

    // Store D: VGPR r of each acc holds (row tile_base + r + 8*khalf, col nt*16 + mlane)
    const int rofs = khalf << 3;
    #pragma unroll
    for (int r = 0; r < 8; ++r) {
        int rowA = rowbase + rofs + r;
        if (rowA < n) {
            float* o = out + (size_t)rowA * COUTC + mlane;
            o[ 0] = accA0[r];
            o[16] = accA1[r];
            o[32] = accA2[r];
            o[48] = accA3[r];
        }
        int rowB = rowbase + 16 + rofs + r;
        if (rowB < n) {
            float* o = out + (size_t)rowB * COUTC + mlane;
            o[ 0] = accB0[r];
            o[16] = accB1[r];
            o[32] = accB2[r];
            o[48] = accB3[r];
        }
    }
}

extern "C" void kernel_launch(void* const* d_in, const int* in_sizes, int n_in,
                              void* d_out, int out_size, void* d_ws, size_t ws_size,
                              hipStream_t stream) {
    const float* feats = (const float*)d_in[0];
    const float* w1    = (const float*)d_in[1];
    const float* w2    = (const float*)d_in[2];
    const int*   coors = (const int*)d_in[3];

    const int n = in_sizes[0] / CINC;   // 400000

    // Workspace layout: grid | nbr | h
    char* ws = (char*)d_ws;
    int* grid = (int*)ws;
    size_t goff = (size_t)BSZ * HDIM * WDIM * sizeof(int);          // 4 MB
    int* nbr = (int*)(ws + goff);
    size_t noff = goff + (((size_t)NTAP * n * sizeof(int) + 255) & ~(size_t)255);
    float* h = (float*)(ws + noff);                                  // n*64 f32

    const int total = BSZ * HDIM * WDIM;
    init_grid_kernel<<<(total + 255) / 256, 256, 0, stream>>>(grid, total);
    scatter_grid_kernel<<<(n + 255) / 256, 256, 0, stream>>>(coors, grid, n);
    build_nbr_kernel<<<(n + 255) / 256, 256, 0, stream>>>(coors, grid, nbr, n);

    const int blocks = (n + 255) / 256;   // 1563
    subm_conv_wmma<<<blocks, 256, 0, stream>>>(feats, w1, nbr, h, n);
    subm_conv_wmma<<<blocks, 256, 0, stream>>>(h, w2, nbr, (float*)d_out, n);
}